// EnhancedFreqLCBlock_53034256171070
// MI455X (gfx1250) — compile-verified
//
#include <hip/hip_runtime.h>
#include <math.h>

typedef float v2f __attribute__((ext_vector_type(2)));
typedef float v8f __attribute__((ext_vector_type(8)));

constexpr int Bb = 2, Cc = 96, Hh = 128, Ww = 128, HW = Hh * Ww;
constexpr int NQ = 4, LQ = 4096, DI = 192, XZn = 384;
constexpr int DBLP = 48, DBLR = 38, DSt = 16, DTR = 6;

__device__ __forceinline__ float silu_f(float x) { return x / (1.f + __expf(-x)); }

// ---------------------------------------------------------------------------
// Wave-level fp32 WMMA tiles (V_WMMA_F32_16X16X4_F32).
// ISA layouts (CDNA5 ISA):
//   A 16x4:  lanes 0-15 hold rows M=0..15 {K=0,K=1}; lanes 16-31 {K=2,K=3}
//   B 4x16:  lanes 0-15 hold cols N=0..15 {K=0,K=1}; lanes 16-31 {K=2,K=3}
//   C/D:     VGPR r, lanes 0-15 -> (M=r, N=lane); lanes 16-31 -> (M=r+8)
// Multi-tile variants amortize the shared operand fragment across several
// accumulators (register-level reuse; GEMMs here are L2-resident and
// load-issue bound, not HBM bound).
// ---------------------------------------------------------------------------

// MT row-tiles (16*MT x 16 output) sharing the B fragment each K-step.
template <int K, int MT, class FA, class FB, class FC>
__device__ __forceinline__ void wmma_tile_m(FA la, FB lb, FC st) {
  const int lane = threadIdx.x & 31;
  const int half = lane >> 4;
  const int l16  = lane & 15;
  v8f acc[MT];
#pragma unroll
  for (int i = 0; i < MT; ++i) acc[i] = v8f{};
#pragma unroll
  for (int k0 = 0; k0 < K; k0 += 4) {
    const int ka = k0 + 2 * half;
    v2f b;
    b.x = lb(ka, l16);
    b.y = lb(ka + 1, l16);
#pragma unroll
    for (int i = 0; i < MT; ++i) {
      v2f a;
      a.x = la(i * 16 + l16, ka);
      a.y = la(i * 16 + l16, ka + 1);
      acc[i] = __builtin_amdgcn_wmma_f32_16x16x4_f32(false, a, false, b,
                                                     (short)0, acc[i],
                                                     false, false);
    }
  }
#pragma unroll
  for (int i = 0; i < MT; ++i)
#pragma unroll
    for (int r = 0; r < 8; ++r) st(i * 16 + r + 8 * half, l16, acc[i][r]);
}

// NT column-tiles (16 x 16*NT output) sharing the A fragment each K-step.
template <int K, int NT, class FA, class FB, class FC>
__device__ __forceinline__ void wmma_tile_n(FA la, FB lb, FC st) {
  const int lane = threadIdx.x & 31;
  const int half = lane >> 4;
  const int l16  = lane & 15;
  v8f acc[NT];
#pragma unroll
  for (int j = 0; j < NT; ++j) acc[j] = v8f{};
#pragma unroll
  for (int k0 = 0; k0 < K; k0 += 4) {
    const int ka = k0 + 2 * half;
    v2f a;
    a.x = la(l16, ka);
    a.y = la(l16, ka + 1);
#pragma unroll
    for (int j = 0; j < NT; ++j) {
      v2f b;
      b.x = lb(ka, j * 16 + l16);
      b.y = lb(ka + 1, j * 16 + l16);
      acc[j] = __builtin_amdgcn_wmma_f32_16x16x4_f32(false, a, false, b,
                                                     (short)0, acc[j],
                                                     false, false);
    }
  }
#pragma unroll
  for (int j = 0; j < NT; ++j)
#pragma unroll
    for (int r = 0; r < 8; ++r) st(r + 8 * half, j * 16 + l16, acc[j][r]);
}

// ------------------------- DCT matrix generation ---------------------------
__global__ void k_dct_init(float* __restrict__ M) {
  int idx = blockIdx.x * 256 + threadIdx.x;       // 128*128
  int i = idx >> 7, j = idx & 127;
  double v = cos(3.14159265358979323846 * (2.0 * j + 1.0) * i / 256.0) *
             sqrt(2.0 / 128.0);
  if (i == 0) v *= 0.7071067811865475244;
  M[idx] = (float)v;
}

// ----------------------- center-similarity masking -------------------------
__global__ void k_mask(const float* __restrict__ x, float* __restrict__ xm) {
  int idx = blockIdx.x * 256 + threadIdx.x;       // B*HW
  int b = idx / HW, hw = idx % HW;
  const float* xb = x + (size_t)b * Cc * HW;
  const int coff = (Hh / 2) * Ww + (Ww / 2);
  float dot = 0.f, nx = 0.f, nc = 0.f;
  for (int c = 0; c < Cc; ++c) {
    float xc = xb[c * HW + coff];
    float xv = xb[c * HW + hw];
    dot += xv * xc; nx += xv * xv; nc += xc * xc;
  }
  float sim = dot / (sqrtf(nc) * sqrtf(nx) + 1e-6f);
  float keep = (sim >= 0.7f) ? 1.f : 0.f;
  float* xo = xm + (size_t)b * Cc * HW;
  for (int c = 0; c < Cc; ++c) xo[c * HW + hw] = xb[c * HW + hw] * keep;
}

// --------------------------- forward DCT (2 GEMMs) -------------------------
// 192 (b,c) slices, each 128x128x128. Wave = 64x16 tile (multi-M x4).
__global__ void __launch_bounds__(256)
k_dct_fwd1(const float* __restrict__ M, const float* __restrict__ X,
           float* __restrict__ T) {
  int w = (blockIdx.x * 256 + threadIdx.x) >> 5;
  int bc = w >> 4, t = w & 15;                    // 2 Msuper * 8 Ntiles
  int r0 = (t >> 3) * 64, c0 = (t & 7) * 16;
  const float* Xb = X + (size_t)bc * HW;
  float* Tb = T + (size_t)bc * HW;
  wmma_tile_m<128, 4>(
      [&](int m, int k) { return M[(r0 + m) * Ww + k]; },
      [&](int k, int n) { return Xb[k * Ww + c0 + n]; },
      [&](int m, int n, float v) { Tb[(r0 + m) * Ww + c0 + n] = v; });
}

__global__ void __launch_bounds__(256)
k_dct_fwd2(const float* __restrict__ M, const float* __restrict__ T,
           float* __restrict__ Y) {
  int w = (blockIdx.x * 256 + threadIdx.x) >> 5;
  int bc = w >> 4, t = w & 15;
  int r0 = (t >> 3) * 64, c0 = (t & 7) * 16;
  const float* Tb = T + (size_t)bc * HW;
  float* Yb = Y + (size_t)bc * HW;
  wmma_tile_m<128, 4>(
      [&](int m, int k) { return Tb[(r0 + m) * Ww + k]; },
      [&](int k, int n) { return M[(c0 + n) * Ww + k]; },   // Mw^T
      [&](int m, int n, float v) { Yb[(r0 + m) * Ww + c0 + n] = v; });
}

// ----------------------------- channel LayerNorm ---------------------------
__global__ void k_ln(const float* __restrict__ xd, const float* __restrict__ lw,
                     const float* __restrict__ lb, float* __restrict__ xn) {
  int idx = blockIdx.x * 256 + threadIdx.x;       // B*HW
  int b = idx / HW, hw = idx % HW;
  const float* xb = xd + (size_t)b * Cc * HW + hw;
  float s = 0.f, s2 = 0.f;
  for (int c = 0; c < Cc; ++c) { float v = xb[c * HW]; s += v; s2 += v * v; }
  float mu = s * (1.f / Cc);
  float var = s2 * (1.f / Cc) - mu * mu;
  float rstd = rsqrtf(var + 1e-5f);
  float* xo = xn + (size_t)b * Cc * HW + hw;
  for (int c = 0; c < Cc; ++c)
    xo[c * HW] = (xb[c * HW] - mu) * rstd * lw[c] + lb[c];
}

// --------------------- in-projection GEMM (quad gather) --------------------
// A is channel-gathered (strided in K) -> multi-N x4 shares the A fragment;
// B (in_w) is contiguous in K -> b64 per tile.
__global__ void __launch_bounds__(256)
k_inproj(const float* __restrict__ xn, const float* __restrict__ in_w,
         float* __restrict__ xz) {
  int w = (blockIdx.x * 256 + threadIdx.x) >> 5;
  int bq = w / 1536, t = w % 1536;                // 256 Mtiles * 6 Nsuper
  int tm = t / 6, ns = t % 6;
  int b = bq >> 2, q = bq & 3;
  int h0 = (q >> 1) * 64, w0 = (q & 1) * 64;
  int r0 = tm * 16, c0 = ns * 64;
  const float* xnb = xn + (size_t)b * Cc * HW;
  const float* wq = in_w + (size_t)q * XZn * Cc;
  float* xzb = xz + ((size_t)bq * LQ + r0) * XZn;
  wmma_tile_n<96, 4>(
      [&](int m, int k) {
        int l = r0 + m;
        return xnb[(size_t)k * HW + (h0 + (l >> 6)) * Ww + (w0 + (l & 63))];
      },
      [&](int k, int n) { return wq[(size_t)(c0 + n) * Cc + k]; },
      [&](int m, int n, float v) { xzb[(size_t)m * XZn + c0 + n] = v; });
}

// ---------------------- depthwise causal conv + SiLU -----------------------
__global__ void k_conv(const float* __restrict__ xz, const float* __restrict__ cw,
                       const float* __restrict__ cb, float* __restrict__ xi) {
  int idx = blockIdx.x * 256 + threadIdx.x;       // B*NQ*LQ*DI
  int d = idx % DI;
  int l = (idx / DI) % LQ;
  int bq = idx / (DI * LQ);
  int q = bq & 3;
  float acc = cb[q * DI + d];
  const float* wv = cw + ((size_t)q * DI + d) * 4;
  const float* xp = xz + (size_t)bq * LQ * XZn + d;
#pragma unroll
  for (int j = 0; j < 4; ++j) {
    int ll = l - 3 + j;
    if (ll >= 0) acc += wv[j] * xp[(size_t)ll * XZn];
  }
  xi[idx] = silu_f(acc);
}

// ------------------------- x-projection GEMM (N=38→48) ---------------------
__global__ void __launch_bounds__(256)
k_xproj(const float* __restrict__ xi, const float* __restrict__ xp_w,
        float* __restrict__ dbl) {
  int w = (blockIdx.x * 256 + threadIdx.x) >> 5;
  int bq = w / 192, t = w % 192;                  // 64 Msuper * 3 Ntiles
  int ms = t / 3, tn = t % 3;
  int q = bq & 3;
  int r0 = ms * 64, c0 = tn * 16;
  const float* Ab = xi + ((size_t)bq * LQ + r0) * DI;
  const float* wq = xp_w + (size_t)q * DBLR * DI;
  float* Db = dbl + ((size_t)bq * LQ + r0) * DBLP;
  wmma_tile_m<192, 4>(
      [&](int m, int k) { return Ab[(size_t)m * DI + k]; },
      [&](int k, int n) {
        int col = c0 + n;
        return (col < DBLR) ? wq[(size_t)col * DI + k] : 0.f;
      },
      [&](int m, int n, float v) { Db[(size_t)m * DBLP + c0 + n] = v; });
}

// ------------------------------ delta (softplus) ---------------------------
__global__ void k_delta(const float* __restrict__ dbl, const float* __restrict__ dtw,
                        const float* __restrict__ dtb, float* __restrict__ dlt) {
  int idx = blockIdx.x * 256 + threadIdx.x;
  int d = idx % DI;
  int l = (idx / DI) % LQ;
  int bq = idx / (DI * LQ);
  int q = bq & 3;
  const float* r = dbl + ((size_t)bq * LQ + l) * DBLP;
  const float* wv = dtw + ((size_t)q * DI + d) * DTR;
  float acc = dtb[q * DI + d];
#pragma unroll
  for (int j = 0; j < DTR; ++j) acc += r[j] * wv[j];
  dlt[idx] = (acc > 0.f) ? (acc + log1pf(__expf(-acc))) : log1pf(__expf(acc));
}

// ------------------- selective scan: 16 lanes = 16 states ------------------
// wave32 = 2 channels; y = sum_s h[s]*C[s] via shfl_xor tree inside 16-group
__global__ void k_scan(const float* __restrict__ dlt, const float* __restrict__ xi,
                       const float* __restrict__ dbl, const float* __restrict__ A_log,
                       float* __restrict__ ysc) {
  int tid = blockIdx.x * 256 + threadIdx.x;       // 24576 lanes = 768 waves
  int lane = tid & 31;
  int s = lane & 15;
  int dsub = lane >> 4;
  int wg = tid >> 5;
  int d = (wg % 96) * 2 + dsub;
  int bq = wg / 96;
  int q = bq & 3;
  float Av = -__expf(A_log[((size_t)q * DI + d) * DSt + s]);
  float h = 0.f;
  for (int l = 0; l < LQ; ++l) {
    size_t base = (size_t)bq * LQ + l;
    float dv = dlt[base * DI + d];
    float xv = xi[base * DI + d];
    float Bv = dbl[base * DBLP + DTR + s];
    float Cv = dbl[base * DBLP + DTR + DSt + s];
    h = __expf(dv * Av) * h + dv * Bv * xv;
    float p = h * Cv;
    p += __shfl_xor(p, 1, 32);
    p += __shfl_xor(p, 2, 32);
    p += __shfl_xor(p, 4, 32);
    p += __shfl_xor(p, 8, 32);
    if (s == 0) ysc[base * DI + d] = p;
  }
}

// ------------------------------- gating ------------------------------------
__global__ void k_gate(const float* __restrict__ ysc, const float* __restrict__ xi,
                       const float* __restrict__ xz, const float* __restrict__ Dp,
                       float* __restrict__ yg) {
  int idx = blockIdx.x * 256 + threadIdx.x;
  int d = idx % DI;
  int l = (idx / DI) % LQ;
  int bq = idx / (DI * LQ);
  int q = bq & 3;
  size_t base = (size_t)bq * LQ + l;
  float yv = ysc[idx] + xi[idx] * Dp[q * DI + d];
  float z = xz[base * XZn + DI + d];
  yg[idx] = yv * silu_f(z);
}

// --------------------------- out-projection GEMM ---------------------------
__global__ void __launch_bounds__(256)
k_outproj(const float* __restrict__ yg, const float* __restrict__ out_w,
          float* __restrict__ qo) {
  int w = (blockIdx.x * 256 + threadIdx.x) >> 5;
  int bq = w / 384, t = w % 384;                  // 64 Msuper * 6 Ntiles
  int ms = t / 6, tn = t % 6;
  int q = bq & 3;
  int r0 = ms * 64, c0 = tn * 16;
  const float* Ab = yg + ((size_t)bq * LQ + r0) * DI;
  const float* wq = out_w + (size_t)q * Cc * DI;
  float* Ob = qo + ((size_t)bq * LQ + r0) * Cc;
  wmma_tile_m<192, 4>(
      [&](int m, int k) { return Ab[(size_t)m * DI + k]; },
      [&](int k, int n) { return wq[(size_t)(c0 + n) * DI + k]; },
      [&](int m, int n, float v) { Ob[(size_t)m * Cc + c0 + n] = v; });
}

// ---------------------- residual add into DCT domain -----------------------
__global__ void k_resid(const float* __restrict__ qo, float* __restrict__ xd) {
  int idx = blockIdx.x * 256 + threadIdx.x;       // B*C*HW
  int b = idx / (Cc * HW);
  int c = (idx / HW) % Cc;
  int hw = idx % HW;
  int h = hw >> 7, ww = hw & 127;
  int q = ((h >> 6) << 1) | (ww >> 6);
  int l = ((h & 63) << 6) | (ww & 63);
  xd[idx] += qo[(((size_t)(b * 4 + q)) * LQ + l) * Cc + c];
}

// ------------------------ inverse DCT (2 GEMMs) ----------------------------
__global__ void __launch_bounds__(256)
k_idct1(const float* __restrict__ M, const float* __restrict__ X,
        float* __restrict__ T) {
  int w = (blockIdx.x * 256 + threadIdx.x) >> 5;
  int bc = w >> 4, t = w & 15;
  int r0 = (t >> 3) * 64, c0 = (t & 7) * 16;
  const float* Xb = X + (size_t)bc * HW;
  float* Tb = T + (size_t)bc * HW;
  wmma_tile_m<128, 4>(
      [&](int m, int k) { return M[k * Ww + r0 + m]; },      // Mh^T
      [&](int k, int n) { return Xb[k * Ww + c0 + n]; },
      [&](int m, int n, float v) { Tb[(r0 + m) * Ww + c0 + n] = v; });
}

__global__ void __launch_bounds__(256)
k_idct2(const float* __restrict__ M, const float* __restrict__ T,
        float* __restrict__ Y) {
  int w = (blockIdx.x * 256 + threadIdx.x) >> 5;
  int bc = w >> 4, t = w & 15;
  int r0 = (t >> 3) * 64, c0 = (t & 7) * 16;
  const float* Tb = T + (size_t)bc * HW;
  float* Yb = Y + (size_t)bc * HW;
  wmma_tile_m<128, 4>(
      [&](int m, int k) { return Tb[(r0 + m) * Ww + k]; },
      [&](int k, int n) { return M[k * Ww + c0 + n]; },      // Mw
      [&](int m, int n, float v) { Yb[(r0 + m) * Ww + c0 + n] = v; });
}

// ---------------------------------------------------------------------------
extern "C" void kernel_launch(void* const* d_in, const int* in_sizes, int n_in,
                              void* d_out, int out_size, void* d_ws, size_t ws_size,
                              hipStream_t stream) {
  (void)in_sizes; (void)n_in; (void)out_size; (void)ws_size;
  const float* x     = (const float*)d_in[0];
  const float* ln_w  = (const float*)d_in[1];
  const float* ln_b  = (const float*)d_in[2];
  const float* in_w  = (const float*)d_in[3];
  const float* cw    = (const float*)d_in[4];
  const float* cb    = (const float*)d_in[5];
  const float* xp_w  = (const float*)d_in[6];
  const float* dt_w  = (const float*)d_in[7];
  const float* dt_b  = (const float*)d_in[8];
  const float* A_log = (const float*)d_in[9];
  const float* Dp    = (const float*)d_in[10];
  const float* out_w = (const float*)d_in[11];
  float* out = (float*)d_out;

  float* ws = (float*)d_ws;
  size_t o = 0;
  float* dct = ws + o;  o += (size_t)128 * 128;
  float* xm  = ws + o;  o += (size_t)Bb * Cc * HW;
  float* t1  = ws + o;  o += (size_t)Bb * Cc * HW;
  float* xd  = ws + o;  o += (size_t)Bb * Cc * HW;
  float* xn  = ws + o;  o += (size_t)Bb * Cc * HW;
  float* xz  = ws + o;  o += (size_t)Bb * NQ * LQ * XZn;
  float* xi  = ws + o;  o += (size_t)Bb * NQ * LQ * DI;
  float* dbl = ws + o;  o += (size_t)Bb * NQ * LQ * DBLP;
  float* dlt = ws + o;  o += (size_t)Bb * NQ * LQ * DI;
  float* ysc = ws + o;  o += (size_t)Bb * NQ * LQ * DI;
  float* yg  = ws + o;  o += (size_t)Bb * NQ * LQ * DI;
  float* qo  = ws + o;  o += (size_t)Bb * NQ * LQ * Cc;
  float* t2  = xm;      // xm dead after forward DCT stage 1 -> reuse for iDCT

  k_dct_init<<<64, 256, 0, stream>>>(dct);
  k_mask<<<128, 256, 0, stream>>>(x, xm);
  k_dct_fwd1<<<384, 256, 0, stream>>>(dct, xm, t1);
  k_dct_fwd2<<<384, 256, 0, stream>>>(dct, t1, xd);
  k_ln<<<128, 256, 0, stream>>>(xd, ln_w, ln_b, xn);
  k_inproj<<<1536, 256, 0, stream>>>(xn, in_w, xz);
  k_conv<<<24576, 256, 0, stream>>>(xz, cw, cb, xi);
  k_xproj<<<192, 256, 0, stream>>>(xi, xp_w, dbl);
  k_delta<<<24576, 256, 0, stream>>>(dbl, dt_w, dt_b, dlt);
  k_scan<<<96, 256, 0, stream>>>(dlt, xi, dbl, A_log, ysc);
  k_gate<<<24576, 256, 0, stream>>>(ysc, xi, xz, Dp, yg);
  k_outproj<<<384, 256, 0, stream>>>(yg, out_w, qo);
  k_resid<<<12288, 256, 0, stream>>>(qo, xd);
  k_idct1<<<384, 256, 0, stream>>>(dct, xd, t2);
  k_idct2<<<384, 256, 0, stream>>>(dct, t2, out);
}